// PConvLinearOpt_8778913153257
// MI455X (gfx1250) — compile-verified
//
#include <hip/hip_runtime.h>
#include <stdint.h>

typedef float v2f __attribute__((ext_vector_type(2)));
typedef float v8f __attribute__((ext_vector_type(8)));

// Problem constants from the reference (fixed sizes)
#define PK    16      // neighbors
#define CIN   64      // input channels
#define CADD  3
#define CTOT  67      // CIN + CADD
#define CPAD  80      // padded channel dim (5 ctiles of 16)
#define MM    16      // weightnet M
#define FDIM  1072    // CTOT * MM
#define COUT  128
#define PSTRIDE 1076  // lds pconv row stride (floats): 1076 mod 64 = 52 -> conflict-free columns
#define FEATS_FLOATS (16 * PK * CPAD)              // 20480
#define PCONV_FLOATS (16 * PSTRIDE)                // 17216
#define DUMP_OFF     (FEATS_FLOATS + PCONV_FLOATS) // scratch row for c>=67 garbage
#define SMEM_FLOATS  (DUMP_OFF + 16)

// One 16x16 pconv output tile for point p, channel-tile ctile:
// D[c_local, m] = sum_{k=0..15} feats[p][k][ctile*16+c_local] * wn[k][m]
// via 4 chained V_WMMA_F32_16X16X4_F32.
__device__ __forceinline__ v8f pconv_tile(const float* __restrict__ lds_feats,
                                          const float* __restrict__ wn,
                                          int p, int ctile, int half, int l16) {
    v8f acc = {};
    #pragma unroll
    for (int k0 = 0; k0 < PK; k0 += 4) {
        const int kr = k0 + half * 2;
        // A fragment: A[c][k] = feats[p][k][c] (M = c, K = k)
        const float* fr = lds_feats + (p * PK + kr) * CPAD + ctile * 16 + l16;
        v2f a; a.x = fr[0]; a.y = fr[CPAD];
        // B fragment: B[k][m] = wn[k][m] (K = k, N = m)
        const float* br = wn + kr * MM + l16;
        v2f bb; bb.x = br[0]; bb.y = br[MM];
        acc = __builtin_amdgcn_wmma_f32_16x16x4_f32(
            false, a, false, bb, (short)0, acc, false, false);
    }
    return acc;
}

__global__ __launch_bounds__(256)
void pconv_linear_fused_kernel(
    const float* __restrict__ in_feat,   // [B,N,64]
    const int*   __restrict__ nbr_inds,  // [B,N,16]
    const float* __restrict__ weightnet, // [B,N,16,16]
    const float* __restrict__ add_feat,  // [B,N,16,3]
    const float* __restrict__ W,         // [128,1072]
    const float* __restrict__ bias,      // [128]
    float*       __restrict__ out,       // [B,N,128]
    int N)
{
    extern __shared__ float smem[];
    float* lds_feats = smem;                       // [16][16][CPAD]
    float* lds_pconv = smem + FEATS_FLOATS;        // [16][PSTRIDE]

    const int tilesPerB = N >> 4;
    const int tile = blockIdx.x;
    const int b  = tile / tilesPerB;
    const int n0 = (tile % tilesPerB) << 4;

    const int t    = threadIdx.x;
    const int lane = t & 31;
    const int wave = t >> 5;           // 0..7
    const int half = lane >> 4;        // 0/1 : selects K pair per WMMA layout
    const int l16  = lane & 15;

    const long long bn0 = (long long)b * N + n0;

    // ---- Stage feats[p][k][c] into LDS: one (p,k) per thread.
    // Gathered 64-ch rows go via async global->LDS DMA (ASYNCcnt-tracked).
    {
        const int p = t >> 4;
        const int k = t & 15;
        const long long bn = bn0 + p;
        const int nbr = nbr_inds[bn * PK + k];
        const float* src = in_feat + ((long long)b * N + nbr) * CIN;
        const unsigned int ldsbase = (unsigned int)((p * PK + k) * CPAD) * 4u; // byte offset
        #pragma unroll
        for (int c = 0; c < CIN; c += 4) {
            unsigned long long ga = (unsigned long long)(uintptr_t)(src + c);
            unsigned int la = ldsbase + (unsigned int)c * 4u;
            asm volatile("global_load_async_to_lds_b128 %0, %1, off"
                         :: "v"(la), "v"(ga) : "memory");
        }
        float* dst = lds_feats + (p * PK + k) * CPAD;
        const float* af = add_feat + (bn * PK + k) * CADD;
        dst[64] = af[0]; dst[65] = af[1]; dst[66] = af[2];
        #pragma unroll
        for (int c = CTOT; c < CPAD; ++c) dst[c] = 0.0f;
        asm volatile("s_wait_asynccnt 0x0" ::: "memory");
    }
    __syncthreads();

    // ---- Phase 1a: the 64 jobs with ctile 0..3 (stores always in-range).
    // Fixed trip count per wave -> scalar loop, EXEC stays untouched.
    #pragma unroll 1
    for (int i = 0; i < 8; ++i) {
        const int job = wave + i * 8;      // 0..63
        const int p = job >> 2;
        const int ctile = job & 3;
        const float* wn = weightnet + (bn0 + p) * (PK * MM);
        v8f acc = pconv_tile(lds_feats, wn, p, ctile, half, l16);
        // D: VGPR j holds row c_local = j + half*8, col m = l16
        float* prow = lds_pconv + p * PSTRIDE + (ctile * 16 + half * 8) * MM + l16;
        #pragma unroll
        for (int j = 0; j < 8; ++j) prow[j * MM] = acc[j];
    }
    // ---- Phase 1b: the 16 jobs with ctile == 4 (rows 64..79; keep 64..66).
    #pragma unroll 1
    for (int i = 0; i < 2; ++i) {
        const int p = wave + i * 8;
        const float* wn = weightnet + (bn0 + p) * (PK * MM);
        v8f acc = pconv_tile(lds_feats, wn, p, /*ctile=*/4, half, l16);
        #pragma unroll
        for (int j = 0; j < 8; ++j) {
            const int c = 64 + half * 8 + j;
            // address select (v_cndmask), no EXEC divergence
            const int addr = (c < CTOT) ? (FEATS_FLOATS + p * PSTRIDE + c * MM + l16)
                                        : (DUMP_OFF + l16);
            smem[addr] = acc[j];
        }
    }
    __syncthreads();

    // ---- Phase 2: out[16 x 128] = pconv[16 x 1072] @ W^T + bias
    // Each wave owns one 16-wide output tile; 67 chunks x 4 WMMA accumulate.
    const int o = wave * 16 + l16;
    v8f acc = {};
    #pragma unroll 1
    for (int c = 0; c < CTOT; ++c) {
        const int fb = c * MM;
        #pragma unroll
        for (int kk = 0; kk < 4; ++kk) {
            const int f0 = fb + kk * 4 + half * 2;
            // A: M = point (l16), K = f
            const float* ar = lds_pconv + l16 * PSTRIDE + f0;
            v2f a; a.x = ar[0]; a.y = ar[1];
            // B: K = f, N = o ; B[f][o] = W[o][f]
            const float* wr = W + (long long)o * FDIM + f0;
            v2f bb; bb.x = wr[0]; bb.y = wr[1];
            acc = __builtin_amdgcn_wmma_f32_16x16x4_f32(
                false, a, false, bb, (short)0, acc, false, false);
        }
    }
    const float bv = bias[o];
    float* orow = out + bn0 * COUT;
    #pragma unroll
    for (int j = 0; j < 8; ++j) {
        const int p = j + half * 8;                // C/D row layout
        orow[(long long)p * COUT + o] = acc[j] + bv;
    }
}

extern "C" void kernel_launch(void* const* d_in, const int* in_sizes, int n_in,
                              void* d_out, int out_size, void* d_ws, size_t ws_size,
                              hipStream_t stream) {
    const float* in_feat   = (const float*)d_in[0];  // input_features [B,N,64]
    const int*   nbr_inds  = (const int*)  d_in[1];  // neighbor_inds  [B,N,16]
    // d_in[2..4]: inverse_* (unused by forward)
    const float* weightnet = (const float*)d_in[5];  // [B,N,16,16]
    const float* add_feat  = (const float*)d_in[6];  // [B,N,16,3]
    const float* W         = (const float*)d_in[7];  // [128,1072]
    const float* bias      = (const float*)d_in[8];  // [128]
    float* out = (float*)d_out;

    const int BN = in_sizes[0] / CIN;   // B*N total points
    const int B  = 2;                   // fixed by reference
    const int N  = BN / B;

    const size_t smem_bytes = (size_t)SMEM_FLOATS * sizeof(float); // ~151 KB
    hipFuncSetAttribute((const void*)pconv_linear_fused_kernel,
                        hipFuncAttributeMaxDynamicSharedMemorySize, (int)smem_bytes);

    const int grid = BN / 16;           // 5000 blocks of 16 points
    pconv_linear_fused_kernel<<<grid, 256, smem_bytes, stream>>>(
        in_feat, nbr_inds, weightnet, add_feat, W, bias, out, N);
}